// TokenInterNeck1_10548439679314
// MI455X (gfx1250) — compile-verified
//
#include <hip/hip_runtime.h>
#include <stdint.h>

// Problem constants (from reference setup_inputs / SCALE_FACTOR=2)
#define BATCH 8
#define NTOK  3072
#define CH    256
#define HH    128           // h*2
#define WW    96            // w*2
#define EPSF  1e-6f

// smooth-kernel tiling
#define TX    32            // x tile
#define TY    8             // y tile
#define TC    32            // channel tile
#define LRX   (TX + 2)      // 34 (x with halo)
#define LRY   (TY + 2)      // 10 (y with halo)
#define NPIX  (LRX * LRY)   // 340 pixels per tile
#define PSTR  33            // LDS dwords per pixel: 32 data + 1 TDM pad (odd -> no bank conflicts)
#define ROWSTR (LRX * PSTR) // 1122 dwords per halo row

typedef unsigned int u32x4 __attribute__((ext_vector_type(4)));
typedef int          i32x8 __attribute__((ext_vector_type(8)));
typedef int          i32x4 __attribute__((ext_vector_type(4)));

// ---------------------------------------------------------------------------
// 1) zero accumulator + count workspace (float4 grid-stride)
// ---------------------------------------------------------------------------
__global__ void zero_kernel(float4* __restrict__ p, size_t n4) {
  size_t i = (size_t)blockIdx.x * blockDim.x + threadIdx.x;
  size_t stride = (size_t)gridDim.x * blockDim.x;
  float4 z = {0.f, 0.f, 0.f, 0.f};
  for (; i < n4; i += stride) p[i] = z;
}

// ---------------------------------------------------------------------------
// 2) scatter tokens into NHWC accumulator grid; one wave32 per token.
//    Lanes sweep channels -> contiguous 128B atomic bursts per wave.
// ---------------------------------------------------------------------------
__global__ __launch_bounds__(256) void scatter_kernel(
    const float* __restrict__ x, const float* __restrict__ loc,
    float* __restrict__ acc, float* __restrict__ cnt) {
  int wave = threadIdx.x >> 5;
  int lane = threadIdx.x & 31;
  int token = blockIdx.x * 8 + wave;
  if (token >= BATCH * NTOK) return;
  int b = token / NTOK;

  float lx = loc[2 * token + 0];
  float ly = loc[2 * token + 1];
  lx = fminf(fmaxf(lx, -1.f), 1.f);
  ly = fminf(fmaxf(ly, -1.f), 1.f);
  float px = 0.5f * (lx + 1.f) * (float)WW - 0.5f;
  float py = 0.5f * (ly + 1.f) * (float)HH - 0.5f;
  int ix = (int)rintf(px);  ix = min(max(ix, 0), WW - 1);   // rintf == RTE == jnp.round
  int iy = (int)rintf(py);  iy = min(max(iy, 0), HH - 1);

  size_t cell = ((size_t)b * HH + iy) * WW + ix;
  const float* xs = x + (size_t)token * CH;
  float* dst = acc + cell * CH;
#pragma unroll
  for (int c = lane; c < CH; c += 32)
    unsafeAtomicAdd(&dst[c], xs[c]);
  if (lane == 0) unsafeAtomicAdd(&cnt[cell], 1.0f);
}

// ---------------------------------------------------------------------------
// TDM issue: one 2D tensor_load_to_lds per halo row.
//   dim0 = 32 contiguous channels (128B line), dim1 = xcount pixels, line
//   stride = CH elements. pad_enable: +1 dword per 32 stored dwords ->
//   LDS pixel stride = 33 dwords.  D# per CDNA5 ISA sec 8.3-8.6.
// ---------------------------------------------------------------------------
__device__ inline void tdm_load_row(const float* gsrc, uint32_t lds_byte_addr,
                                    i32x8 g1) {
  uint64_t ga = (uint64_t)(uintptr_t)gsrc;
  u32x4 g0;
  g0.x = 1u;                                  // count=1, user mode
  g0.y = lds_byte_addr;                       // lds_addr [63:32]
  g0.z = (uint32_t)ga;                        // global_addr[31:0]
  g0.w = (uint32_t)((ga >> 32) & 0x01FFFFFFu) // global_addr[56:32]
         | 0x80000000u;                       // type=2 ("image") [127:126]=10b

  i32x4 zz = {0, 0, 0, 0};
#if __has_include(<hip/amd_detail/amd_gfx1250_TDM.h>)
  i32x8 z8 = {0, 0, 0, 0, 0, 0, 0, 0};
  __builtin_amdgcn_tensor_load_to_lds(g0, g1, zz, zz, z8, 0);
#else
  __builtin_amdgcn_tensor_load_to_lds(g0, g1, zz, zz, 0);
#endif
}

__device__ inline i32x8 tdm_group1(int xcount) {
  i32x8 g1;
  g1[0] = (int)((2u << 16)        // data_size = 4B
              | (1u << 20)        // pad_enable
              | (4u << 22)        // pad_interval code 4 = 32 dwords
              | (0u << 25));      // pad_amount code 0 = 1 dword
  g1[1] = (int)((uint32_t)TC << 16);      // tensor_dim0 lo16 = 32 (== tile_dim0, no OOB)
  g1[2] = (int)((uint32_t)xcount << 16);  // tensor_dim0 hi16=0 | tensor_dim1 lo16 = xcount
  g1[3] = (int)((uint32_t)TC << 16);      // tensor_dim1 hi16=0 | tile_dim0 = 32
  g1[4] = (int)((uint32_t)xcount);        // tile_dim1 = xcount | tile_dim2 = 0 (2D)
  g1[5] = (int)((uint32_t)CH);            // tensor_dim0_stride lo32 = 256 elements
  g1[6] = 0;                              // stride hi16 | tensor_dim1_stride lo16
  g1[7] = 0;                              // tensor_dim1_stride hi32
  return g1;
}

// ---------------------------------------------------------------------------
// 3) fused normalize + 3x3 gaussian smooth + hole-fill + NHWC->NCHW transpose
//    Raw acc tile streamed into LDS by the Tensor Data Mover, overlapped with
//    the count-tile processing (disjoint LDS addresses: TDM writes in-bounds
//    pixels, the zero phase writes only OOB halo pixels).
// ---------------------------------------------------------------------------
__global__ __launch_bounds__(256) void smooth_kernel(
    const float* __restrict__ acc, const float* __restrict__ cnt,
    float* __restrict__ out) {
  __shared__ float rawL[LRY * ROWSTR];  // [ry][rx][c], pixel stride 33 dwords
  __shared__ float scaleL[NPIX];        // per-pixel masked 1/(count+eps)
  __shared__ float maskL[NPIX];         // per-pixel occupancy

  const int tid = threadIdx.x;
  const int b = blockIdx.z >> 3;
  const int cbase = (blockIdx.z & 7) * TC;
  const int x0 = blockIdx.x * TX - 1;   // halo origin
  const int y0 = blockIdx.y * TY - 1;

  // ---- phase 1a: wave 0 issues TDM loads for all in-bounds halo rows ----
  if (tid < 32) {
    int gx0 = x0, rx0 = 0;
    if (gx0 < 0) { gx0 = 0; rx0 = 1; }            // clip left halo
    int xcount = min(x0 + LRX, WW) - gx0;         // clip right halo
    i32x8 g1 = tdm_group1(xcount);                // row-invariant descriptor part
    int ry0 = (y0 < 0) ? 1 : 0;
    int ry1 = min(LRY, HH - y0);
    for (int ry = ry0; ry < ry1; ++ry) {
      const float* gsrc =
          acc + (((size_t)b * HH + (y0 + ry)) * WW + gx0) * CH + cbase;
      uint32_t la = (uint32_t)(uintptr_t)(&rawL[ry * ROWSTR + rx0 * PSTR]);
      tdm_load_row(gsrc, la, g1);
    }
  }

  // ---- phase 1b (overlapped with DMA): scale/mask tiles + zero OOB pixels --
  for (int e = tid; e < NPIX; e += 256) {
    int ry = e / LRX, rx = e - ry * LRX;
    int gy = y0 + ry, gx = x0 + rx;
    bool inb = (unsigned)gy < HH && (unsigned)gx < WW;
    float cv = inb ? cnt[((size_t)b * HH + gy) * WW + gx] : 0.f;
    bool occ = cv > 0.f;
    scaleL[e] = occ ? 1.f / (cv + EPSF) : 0.f;
    maskL[e]  = occ ? 1.f : 0.f;
    if (!inb) {                       // only OOB halo pixels; disjoint from TDM
      float* zp = &rawL[ry * ROWSTR + rx * PSTR];
#pragma unroll
      for (int c = 0; c < TC; ++c) zp[c] = 0.f;
    }
  }

  __builtin_amdgcn_s_wait_tensorcnt(0);           // all waves (no-op if cnt==0)
  __syncthreads();

  // ---- phase 2: compute ----
  // gaussian weights, k=3 sigma=2: exp(-(dx^2+dy^2)/8)/norm
  const float e1 = expf(-0.125f);
  const float e2 = expf(-0.25f);
  const float nrm = 1.f + 4.f * (e1 + e2);
  float wk[3] = {1.f / nrm, e1 / nrm, e2 / nrm};

  const int xt = tid & 31;    // lanes along x -> coalesced NCHW stores
  const int cg = tid >> 5;    // 8 channel groups; each thread does 4 channels

  for (int y = 0; y < TY; ++y) {
    float a9[9];
    float ms = 0.f;
#pragma unroll
    for (int dy = 0; dy < 3; ++dy)
#pragma unroll
      for (int dx = 0; dx < 3; ++dx) {
        int p = (y + dy) * LRX + xt + dx;
        float w = wk[(dy == 1 ? 0 : 1) + (dx == 1 ? 0 : 1)];
        a9[dy * 3 + dx] = w * scaleL[p];   // fused neighbor normalization
        ms += w * maskL[p];
      }
    int pc = (y + 1) * LRX + xt + 1;
    float mc = maskL[pc];
    float invm = (ms > 0.f) ? 1.f / (ms + EPSF) : 0.f;
    float blend = (1.f - mc) * invm;       // (1-mask) * (mask_s>0)/(mask_s+eps)
    float scc = scaleL[pc];
    int pcl = (y + 1) * ROWSTR + (xt + 1) * PSTR;

#pragma unroll
    for (int k = 0; k < 4; ++k) {
      int c = cg + k * 8;
      float fs = 0.f;
#pragma unroll
      for (int dy = 0; dy < 3; ++dy)
#pragma unroll
        for (int dx = 0; dx < 3; ++dx)
          fs += a9[dy * 3 + dx] * rawL[(y + dy) * ROWSTR + (xt + dx) * PSTR + c];
      float fc = rawL[pcl + c] * scc;      // normalized masked center feature
      float val = fc + blend * fs;         // occupied -> fc; empty -> smoothed fill
      out[(((size_t)b * CH + cbase + c) * HH + (y0 + 1 + y)) * WW + (x0 + 1 + xt)] = val;
    }
  }
}

// ---------------------------------------------------------------------------
extern "C" void kernel_launch(void* const* d_in, const int* in_sizes, int n_in,
                              void* d_out, int out_size, void* d_ws, size_t ws_size,
                              hipStream_t stream) {
  const float* x   = (const float*)d_in[0];   // (8, 3072, 256)
  const float* loc = (const float*)d_in[1];   // (8, 3072, 2)
  float* outp = (float*)d_out;                // (8, 256, 128, 96)

  float* acc = (float*)d_ws;                             // B*H*W*C floats
  float* cnt = acc + (size_t)BATCH * HH * WW * CH;       // B*H*W floats

  size_t tot4 = ((size_t)BATCH * HH * WW * (CH + 1)) / 4;     // divisible by 4
  zero_kernel<<<2048, 256, 0, stream>>>((float4*)d_ws, tot4);

  scatter_kernel<<<(BATCH * NTOK) / 8, 256, 0, stream>>>(x, loc, acc, cnt);

  dim3 grid(WW / TX, HH / TY, BATCH * (CH / TC));  // (3, 16, 64)
  smooth_kernel<<<grid, 256, 0, stream>>>(acc, cnt, outp);
}